// GraphPredNetwork_86818468921765
// MI455X (gfx1250) — compile-verified
//
#include <hip/hip_runtime.h>
#include <hip/hip_bf16.h>
#include <math.h>

// ---------------------------------------------------------------------------
// Shapes
// ---------------------------------------------------------------------------
#define BT 16      // B*T = 2*8
#define N  128
#define H  128
#define E  10
#define S  6
#define NPAD 132   // padded hidden-row stride in LDS (avoids bank conflicts)

typedef __attribute__((ext_vector_type(2))) float v2f;
typedef __attribute__((ext_vector_type(8))) float v8f;

__device__ __forceinline__ v8f wmma_f32_16x16x4(v2f a, v2f b, v8f c) {
  // (neg_a, A, neg_b, B, c_mod, C, reuse_a, reuse_b)
  return __builtin_amdgcn_wmma_f32_16x16x4_f32(false, a, false, b, (short)0, c,
                                               false, false);
}

__device__ __forceinline__ float sigm(float x) { return 1.0f / (1.0f + __expf(-x)); }

// ---------------------------------------------------------------------------
// Kernel 1: pooling + comb + 2-layer LSTM + c_bt = h1 @ comb_out_W_top + b
// grid = B (=2) blocks, 128 threads. Tiny & sequential; plain VALU is fine.
// ---------------------------------------------------------------------------
__global__ void lstm_kernel(const float* __restrict__ node_emb,  // [BT][N][H]
                            const float* __restrict__ mask,      // [BT][N]
                            const int*   __restrict__ action,    // [BT]
                            const float* __restrict__ act_table, // [A][H]
                            const float* __restrict__ combW,     // [2H][H]
                            const float* __restrict__ combB,     // [H]
                            const float* __restrict__ Wih0, const float* __restrict__ Whh0,
                            const float* __restrict__ bih0, const float* __restrict__ bhh0,
                            const float* __restrict__ Wih1, const float* __restrict__ Whh1,
                            const float* __restrict__ bih1, const float* __restrict__ bhh1,
                            const float* __restrict__ outW,      // [2H][H] (top used)
                            const float* __restrict__ outB,      // [H]
                            float* __restrict__ cbt)             // [BT][H]
{
  __shared__ float sact[H];
  __shared__ float sgr[H];
  __shared__ float sx[H];
  __shared__ float sh0[H];
  __shared__ float sh1[H];

  const int b = blockIdx.x;        // 0..1
  const int h = threadIdx.x;       // 0..127
  float c0 = 0.f, c1 = 0.f;
  sh0[h] = 0.f; sh1[h] = 0.f;
  __syncthreads();

  for (int t = 0; t < 8; ++t) {
    const int bt = b * 8 + t;
    // masked pool over nodes + action embedding
    float gr = 0.f;
    const float* ne = node_emb + (size_t)bt * N * H;
    const float* mk = mask + (size_t)bt * N;
    for (int n = 0; n < N; ++n) gr += ne[n * H + h] * mk[n];
    sgr[h]  = gr;
    sact[h] = act_table[action[bt] * H + h];
    __syncthreads();

    // input_embed = concat(act, gr) @ combW + combB
    float xh = combB[h];
    for (int k = 0; k < H; ++k) xh += sact[k] * combW[k * H + h];
    for (int k = 0; k < H; ++k) xh += sgr[k] * combW[(H + k) * H + h];
    sx[h] = xh;
    __syncthreads();

    // LSTM layer 0 (gate order i,f,g,o; W stored [in,4H])
    {
      float gi = bih0[h] + bhh0[h];
      float gf = bih0[128 + h] + bhh0[128 + h];
      float gg = bih0[256 + h] + bhh0[256 + h];
      float go = bih0[384 + h] + bhh0[384 + h];
      for (int k = 0; k < H; ++k) {
        const float xv = sx[k], hv = sh0[k];
        gi += xv * Wih0[k * 512 + h]       + hv * Whh0[k * 512 + h];
        gf += xv * Wih0[k * 512 + 128 + h] + hv * Whh0[k * 512 + 128 + h];
        gg += xv * Wih0[k * 512 + 256 + h] + hv * Whh0[k * 512 + 256 + h];
        go += xv * Wih0[k * 512 + 384 + h] + hv * Whh0[k * 512 + 384 + h];
      }
      c0 = sigm(gf) * c0 + sigm(gi) * tanhf(gg);
      const float h0v = sigm(go) * tanhf(c0);
      __syncthreads();
      sh0[h] = h0v;
      __syncthreads();
    }
    // LSTM layer 1
    {
      float gi = bih1[h] + bhh1[h];
      float gf = bih1[128 + h] + bhh1[128 + h];
      float gg = bih1[256 + h] + bhh1[256 + h];
      float go = bih1[384 + h] + bhh1[384 + h];
      for (int k = 0; k < H; ++k) {
        const float xv = sh0[k], hv = sh1[k];
        gi += xv * Wih1[k * 512 + h]       + hv * Whh1[k * 512 + h];
        gf += xv * Wih1[k * 512 + 128 + h] + hv * Whh1[k * 512 + 128 + h];
        gg += xv * Wih1[k * 512 + 256 + h] + hv * Whh1[k * 512 + 256 + h];
        go += xv * Wih1[k * 512 + 384 + h] + hv * Whh1[k * 512 + 384 + h];
      }
      c1 = sigm(gf) * c1 + sigm(gi) * tanhf(gg);
      const float h1v = sigm(go) * tanhf(c1);
      __syncthreads();
      sh1[h] = h1v;
      __syncthreads();
    }
    // c_bt = h1 @ outW_top + outB  (per-(b,t) constant column bias for `out`)
    float ob = outB[h];
    for (int k = 0; k < H; ++k) ob += sh1[k] * outW[k * H + h];
    cbt[bt * H + h] = ob;
    __syncthreads();
  }
}

// ---------------------------------------------------------------------------
// Generic 128x128x128 GEMM per group g:  Out[g] = op(A[g] @ W + bias)
// grid = (8, 8, G), block = 32 (one wave -> one 16x16 tile, K=128 via 32 WMMA)
// f32 WMMA layouts per CDNA5 ISA:
//   A 16x4: lane=16*half+m holds A[m][2*half + {0,1}]
//   B 4x16: lane=16*half+n holds B[2*half + {0,1}][n]
//   C/D:    vgpr r, lane -> D[r + 8*half][lane&15]
// ---------------------------------------------------------------------------
__global__ void gemm128_wmma(const float* __restrict__ A,    // [G][128][128]
                             const float* __restrict__ W,    // [128][128] (row k, col c)
                             const float* __restrict__ bias, // [128] or [G][128] or null
                             int bias_per_g, int do_relu,
                             float* __restrict__ Out)        // [G][128][128]
{
  const int g    = blockIdx.z;
  const int m0   = blockIdx.x * 16;
  const int n0   = blockIdx.y * 16;
  const int lane = threadIdx.x;    // 0..31
  const int half = lane >> 4;
  const int mn   = lane & 15;

  const float* Ag = A + (size_t)g * 128 * 128;
  v8f acc = {};
  for (int k0 = 0; k0 < 128; k0 += 4) {
    const int ka = k0 + half * 2;
    const v2f a = *(const v2f*)(Ag + (m0 + mn) * 128 + ka);   // 8B aligned
    v2f b;
    b.x = W[(ka    ) * 128 + (n0 + mn)];
    b.y = W[(ka + 1) * 128 + (n0 + mn)];
    acc = wmma_f32_16x16x4(a, b, acc);
  }

  float bval = 0.f;
  if (bias) bval = bias_per_g ? bias[g * 128 + n0 + mn] : bias[n0 + mn];

  float* Og = Out + (size_t)g * 128 * 128;
#pragma unroll
  for (int r = 0; r < 8; ++r) {
    const int m = m0 + r + 8 * half;
    float v = acc[r] + bval;
    if (do_relu) v = fmaxf(v, 0.f);
    Og[m * 128 + (n0 + mn)] = v;
  }
}

// ---------------------------------------------------------------------------
// states = Hs @ state_W2 (128x6) + b2 ; tiny
// ---------------------------------------------------------------------------
__global__ void states_kernel(const float* __restrict__ Hs, // [BT*N][128]
                              const float* __restrict__ W2, // [128][6]
                              const float* __restrict__ b2, // [6]
                              float* __restrict__ states)   // [BT*N][6]
{
  const int idx = blockIdx.x * blockDim.x + threadIdx.x;
  if (idx >= BT * N * S) return;
  const int s = idx % S;
  const int r = idx / S;
  float acc = b2[s];
  const float* hrow = Hs + (size_t)r * 128;
  for (int k = 0; k < 128; ++k) acc += hrow[k] * W2[k * S + s];
  states[idx] = acc;
}

// ---------------------------------------------------------------------------
// Pair kernel: for each (bt, i), for all j:
//   he = relu(Pe[j] + Qe[i])  (edge_b1 folded into Pe)
//   edges[i*N+j]  = he @ edge_W2 + edge_b2     (128 -> 10, padded to 16)
//   hc = relu(Pc[j] + Qc[i])
//   change[i*N+j] = hc @ chg_W2 + chg_b2       (128 -> 2,  padded to 16)
// grid = BT*N blocks, 64 threads (2 waves). Each wave builds a 16x128 hidden
// tile in LDS (row stride NPAD=132 -> conflict-free b64 A-fragment reads) and
// runs 2x32 f32 WMMAs against W2 tiles resident in LDS (padded to 16 cols).
// ---------------------------------------------------------------------------
__global__ void pair_kernel(const float* __restrict__ Pe, const float* __restrict__ Qe,
                            const float* __restrict__ Pc, const float* __restrict__ Qc,
                            const float* __restrict__ eW2, const float* __restrict__ eb2,
                            const float* __restrict__ cW2, const float* __restrict__ cb2,
                            float* __restrict__ edges,    // [BT][N*N][10]
                            float* __restrict__ change)   // [BT][N*N][2]
{
  __shared__ float sWe[128 * 16];
  __shared__ float sWc[128 * 16];
  __shared__ float sqe[128];
  __shared__ float sqc[128];
  __shared__ float she[2][16 * NPAD];
  __shared__ float shc[2][16 * NPAD];

  const int g    = blockIdx.x >> 7;   // bt
  const int i    = blockIdx.x & 127;
  const int tid  = threadIdx.x;       // 0..63
  const int w    = tid >> 5;
  const int lane = tid & 31;
  const int half = lane >> 4;
  const int mn   = lane & 15;

  // stage W2 matrices (zero-padded to 16 output cols) and Q rows
  for (int idx = tid; idx < 128 * 16; idx += 64) {
    const int k = idx >> 4, n = idx & 15;
    sWe[idx] = (n < E) ? eW2[k * E + n] : 0.f;
    sWc[idx] = (n < 2) ? cW2[k * 2 + n] : 0.f;
  }
  for (int hh = tid; hh < 128; hh += 64) {
    sqe[hh] = Qe[((size_t)g * N + i) * 128 + hh];
    sqc[hh] = Qc[((size_t)g * N + i) * 128 + hh];
  }
  __syncthreads();

  float* he = &she[w][0];
  float* hc = &shc[w][0];

  for (int t = 0; t < 4; ++t) {
    const int j0 = (w * 4 + t) * 16;   // waves cover j tiles 0..7

    // build relu(P[j]+Q[i]) hidden tiles (coalesced global reads)
    for (int idx = lane; idx < 16 * 128; idx += 32) {
      const int m = idx >> 7, hh = idx & 127;
      const float pe = Pe[((size_t)g * N + (j0 + m)) * 128 + hh];
      const float pc = Pc[((size_t)g * N + (j0 + m)) * 128 + hh];
      he[m * NPAD + hh] = fmaxf(pe + sqe[hh], 0.f);
      hc[m * NPAD + hh] = fmaxf(pc + sqc[hh], 0.f);
    }
    __syncthreads();

    v8f ae = {};
    v8f ac = {};
    for (int k0 = 0; k0 < 128; k0 += 4) {
      const int ka = k0 + half * 2;
      const v2f a_e = *(const v2f*)&he[mn * NPAD + ka];
      v2f b_e; b_e.x = sWe[ka * 16 + mn]; b_e.y = sWe[(ka + 1) * 16 + mn];
      ae = wmma_f32_16x16x4(a_e, b_e, ae);
      const v2f a_c = *(const v2f*)&hc[mn * NPAD + ka];
      v2f b_c; b_c.x = sWc[ka * 16 + mn]; b_c.y = sWc[(ka + 1) * 16 + mn];
      ac = wmma_f32_16x16x4(a_c, b_c, ac);
    }

    // write results (divergence only after all WMMAs; EXEC was all-1s there)
    if (mn < E) {
      const float bb = eb2[mn];
#pragma unroll
      for (int r = 0; r < 8; ++r) {
        const int m = r + 8 * half;
        const int row = i * N + j0 + m;
        edges[((size_t)g * N * N + row) * E + mn] = ae[r] + bb;
      }
    }
    if (mn < 2) {
      const float bb = cb2[mn];
#pragma unroll
      for (int r = 0; r < 8; ++r) {
        const int m = r + 8 * half;
        const int row = i * N + j0 + m;
        change[((size_t)g * N * N + row) * 2 + mn] = ac[r] + bb;
      }
    }
    __syncthreads();   // protect he/hc before next tile
  }
}

// ---------------------------------------------------------------------------
// Launch
// ---------------------------------------------------------------------------
extern "C" void kernel_launch(void* const* d_in, const int* in_sizes, int n_in,
                              void* d_out, int out_size, void* d_ws, size_t ws_size,
                              hipStream_t stream) {
  const float* node_emb  = (const float*)d_in[0];
  const float* mask      = (const float*)d_in[1];
  const int*   action    = (const int*)  d_in[2];
  const float* act_table = (const float*)d_in[3];
  const float* combW     = (const float*)d_in[4];
  const float* combB     = (const float*)d_in[5];
  const float* Wih0      = (const float*)d_in[6];
  const float* Whh0      = (const float*)d_in[7];
  const float* bih0      = (const float*)d_in[8];
  const float* bhh0      = (const float*)d_in[9];
  const float* Wih1      = (const float*)d_in[10];
  const float* Whh1      = (const float*)d_in[11];
  const float* bih1      = (const float*)d_in[12];
  const float* bhh1      = (const float*)d_in[13];
  const float* combOutW  = (const float*)d_in[14];
  const float* combOutB  = (const float*)d_in[15];
  const float* stateW1   = (const float*)d_in[16];
  const float* stateB1   = (const float*)d_in[17];
  const float* stateW2   = (const float*)d_in[18];
  const float* stateB2   = (const float*)d_in[19];
  const float* edgeW1    = (const float*)d_in[20];
  const float* edgeB1    = (const float*)d_in[21];
  const float* edgeW2    = (const float*)d_in[22];
  const float* edgeB2    = (const float*)d_in[23];
  const float* chgW1     = (const float*)d_in[24];
  const float* chgB1     = (const float*)d_in[25];
  const float* chgW2     = (const float*)d_in[26];
  const float* chgB2     = (const float*)d_in[27];

  float* ws  = (float*)d_ws;
  float* cbt = ws;                                  // [16][128]
  float* out = cbt + BT * H;                        // [16][128][128]
  float* Hs  = out + BT * N * H;
  float* Pe  = Hs + BT * N * H;
  float* Qe  = Pe + BT * N * H;
  float* Pc  = Qe + BT * N * H;
  float* Qc  = Pc + BT * N * H;

  float* states = (float*)d_out;                    // [16][128][6]
  float* edges  = states + BT * N * S;              // [16][16384][10]
  float* change = edges + (size_t)BT * N * N * E;   // [16][16384][2]

  // 1) sequential head -> c_bt
  lstm_kernel<<<2, 128, 0, stream>>>(node_emb, mask, action, act_table,
                                     combW, combB,
                                     Wih0, Whh0, bih0, bhh0,
                                     Wih1, Whh1, bih1, bhh1,
                                     combOutW, combOutB, cbt);

  // 2) out = node_emb @ comb_out_W_bot + c_bt   (per-bt column bias)
  dim3 g88(8, 8, BT);
  gemm128_wmma<<<g88, 32, 0, stream>>>(node_emb, combOutW + 128 * 128, cbt, 1, 0, out);

  // 3) Hs = relu(out @ state_W1 + state_b1)
  gemm128_wmma<<<g88, 32, 0, stream>>>(out, stateW1, stateB1, 0, 1, Hs);

  // 4) P/Q factors of the pairwise MLPs (b1 folded into P)
  gemm128_wmma<<<g88, 32, 0, stream>>>(out, edgeW1,             edgeB1, 0, 0, Pe);
  gemm128_wmma<<<g88, 32, 0, stream>>>(out, edgeW1 + 128 * 128, nullptr, 0, 0, Qe);
  gemm128_wmma<<<g88, 32, 0, stream>>>(out, chgW1,              chgB1,  0, 0, Pc);
  gemm128_wmma<<<g88, 32, 0, stream>>>(out, chgW1 + 128 * 128,  nullptr, 0, 0, Qc);

  // 5) states projection (tiny)
  states_kernel<<<(BT * N * S + 255) / 256, 256, 0, stream>>>(Hs, stateW2, stateB2, states);

  // 6) fused all-pairs edges + change
  pair_kernel<<<BT * N, 64, 0, stream>>>(Pe, Qe, Pc, Qc,
                                         edgeW2, edgeB2, chgW2, chgB2,
                                         edges, change);
}